// AtomGCNLayer_19997367730281
// MI455X (gfx1250) — compile-verified
//
#include <hip/hip_runtime.h>
#include <math.h>

#define N_NODES 500000
#define N_EDGES 5000000
#define EMBD 16
#define BN_EPS 1e-5f

#define NODE_TILES (N_NODES / 16)   // 31250
#define EDGE_TILES (N_EDGES / 16)   // 312500
#define WAVES_PER_BLOCK 8

typedef float v2f __attribute__((ext_vector_type(2)));
typedef float v8f __attribute__((ext_vector_type(8)));
typedef int   v4i __attribute__((vector_size(16)));   // matches builtin param type

#define AS1 __attribute__((address_space(1)))
#define AS3 __attribute__((address_space(3)))

// Async global -> LDS 16B copy per lane (gfx1250, tracked on ASYNCcnt).
__device__ __forceinline__ void async_copy_b128(const float* g, float* l) {
    __builtin_amdgcn_global_load_async_to_lds_b128((AS1 v4i*)(g), (AS3 v4i*)(l), 0, 0);
}

// ---------------------------------------------------------------------------
// Phase 1: node-side linears. One wave = one 16-node tile.
// k = x@Wk+bk, q = x@Wq+bq, v = x@Wv+bv -> workspace (L2-resident),
// h = x@Ws+bs+bias -> d_out (scatter-add target for phase 2).
// V_WMMA_F32_16X16X4_F32 chained over 4 K-slices (K=16).
// ---------------------------------------------------------------------------
__global__ void node_linear_kernel(const float* __restrict__ x,
                                   const float* __restrict__ Wk, const float* __restrict__ bk,
                                   const float* __restrict__ Wq, const float* __restrict__ bq,
                                   const float* __restrict__ Wv, const float* __restrict__ bv,
                                   const float* __restrict__ Ws, const float* __restrict__ bs,
                                   const float* __restrict__ bias,
                                   float* __restrict__ kOut, float* __restrict__ qOut,
                                   float* __restrict__ vOut, float* __restrict__ hOut) {
    const int lane = threadIdx.x & 31;
    const int wave = threadIdx.x >> 5;
    const int tile = blockIdx.x * WAVES_PER_BLOCK + wave;
    if (tile >= NODE_TILES) return;          // wave-uniform -> EXEC stays all-ones
    const int m    = lane & 15;              // A: row within tile;  B/C: column
    const int half = lane >> 4;              // 0 or 1

    // A-matrix (x tile) and B-matrices (weights) for 4 K-slices of 4.
    // slice s, vgpr j holds K = 4*s + j + 2*half.
    v2f a[4], Bk_[4], Bq_[4], Bv_[4], Bs_[4];
    const int rowBase = (tile * 16 + m) * EMBD;
#pragma unroll
    for (int s = 0; s < 4; ++s) {
#pragma unroll
        for (int j = 0; j < 2; ++j) {
            const int kk = 4 * s + j + 2 * half;
            a[s][j]   = x [rowBase + kk];
            Bk_[s][j] = Wk[kk * EMBD + m];
            Bq_[s][j] = Wq[kk * EMBD + m];
            Bv_[s][j] = Wv[kk * EMBD + m];
            Bs_[s][j] = Ws[kk * EMBD + m];
        }
    }

    v8f ck = {}; v8f cq = {}; v8f cv = {}; v8f cs = {};
#pragma unroll
    for (int s = 0; s < 4; ++s) {
        ck = __builtin_amdgcn_wmma_f32_16x16x4_f32(false, a[s], false, Bk_[s], (short)0, ck, false, false);
        cq = __builtin_amdgcn_wmma_f32_16x16x4_f32(false, a[s], false, Bq_[s], (short)0, cq, false, false);
        cv = __builtin_amdgcn_wmma_f32_16x16x4_f32(false, a[s], false, Bv_[s], (short)0, cv, false, false);
        cs = __builtin_amdgcn_wmma_f32_16x16x4_f32(false, a[s], false, Bs_[s], (short)0, cs, false, false);
    }

    const float bkv = bk[m], bqv = bq[m], bvv = bv[m];
    const float bsv = bs[m] + bias[m];
    // C/D layout: vgpr i -> row (i + 8*half), lane%16 -> column.
#pragma unroll
    for (int i = 0; i < 8; ++i) {
        const int node = tile * 16 + i + 8 * half;
        const int idx  = node * EMBD + m;
        kOut[idx] = ck[i] + bkv;
        qOut[idx] = cq[i] + bqv;
        vOut[idx] = cv[i] + bvv;
        hOut[idx] = cs[i] + bsv;
    }
}

// ---------------------------------------------------------------------------
// Phase 2: edge GEMM (edge_attr @ We) via WMMA + gather/gate/scatter-add.
// One wave = one 16-edge tile (1 KB). The 320 MB edge_attr stream is staged
// into LDS with coalesced async b128 copies (2 x 512 B per tile, ASYNCcnt),
// then the WMMA A-layout is read back with ds_load_b64.
// ---------------------------------------------------------------------------
__global__ void edge_message_kernel(const float* __restrict__ edge_attr,
                                    const int*  __restrict__ eiSrc,
                                    const int*  __restrict__ eiTgt,
                                    const float* __restrict__ We,
                                    const float* __restrict__ kArr,
                                    const float* __restrict__ qArr,
                                    const float* __restrict__ vArr,
                                    float* __restrict__ hOut) {
    __shared__ float stage[WAVES_PER_BLOCK][16 * EMBD];   // 1 KB per wave

    const int lane = threadIdx.x & 31;
    const int wave = threadIdx.x >> 5;
    const int tile = blockIdx.x * WAVES_PER_BLOCK + wave;
    if (tile >= EDGE_TILES) return;          // wave-uniform
    const int m    = lane & 15;
    const int half = lane >> 4;

    // ---- coalesced async stage of the 16x16 edge_attr tile into LDS ----
    const float* gbase = edge_attr + (size_t)tile * (16 * EMBD);
    float*       lbase = &stage[wave][0];
    async_copy_b128(gbase + lane * 4,       lbase + lane * 4);         // bytes   0..511
    async_copy_b128(gbase + 128 + lane * 4, lbase + 128 + lane * 4);   // bytes 512..1023
    asm volatile("s_wait_asynccnt 0" ::: "memory");

    // ---- B matrix (We) + A matrix from LDS (kk,kk+1 pair is 8B contiguous) ----
    v2f a[4], Be[4];
#pragma unroll
    for (int s = 0; s < 4; ++s) {
        a[s] = *(const v2f*)&stage[wave][m * EMBD + 4 * s + 2 * half]; // ds_load_b64
#pragma unroll
        for (int j = 0; j < 2; ++j) {
            const int kk = 4 * s + j + 2 * half;
            Be[s][j] = We[kk * EMBD + m];
        }
    }

    v8f ce = {};
#pragma unroll
    for (int s = 0; s < 4; ++s) {
        ce = __builtin_amdgcn_wmma_f32_16x16x4_f32(false, a[s], false, Be[s], (short)0, ce, false, false);
    }

    // C layout: vgpr i -> edge (tile*16 + i + 8*half), dim = lane%16.
    // For fixed i each 16-lane half shares one edge -> broadcast index loads,
    // 64B-contiguous k/q/v gathers (L2-resident), 64B-contiguous fp32 atomics.
#pragma unroll
    for (int i = 0; i < 8; ++i) {
        const int edge = tile * 16 + i + 8 * half;
        const int sn   = eiSrc[edge];
        const int tn   = eiTgt[edge];
        const float pre  = kArr[tn * EMBD + m] + ce[i] + qArr[sn * EMBD + m];
        const float gate = 1.0f / (1.0f + __expf(-pre));
        const float msg  = gate * vArr[sn * EMBD + m];
        unsafeAtomicAdd(&hOut[tn * EMBD + m], msg);   // global_atomic_add_f32
    }
}

// ---------------------------------------------------------------------------
// Phase 3: BatchNorm statistics. Per-dim sum / sum-of-squares: LDS ds_add_f32
// block reduction, then 32 global fp32 atomics per block.
// ---------------------------------------------------------------------------
__global__ void bn_stats_kernel(const float* __restrict__ h, float* __restrict__ stats) {
    __shared__ float ssum[EMBD];
    __shared__ float ssq[EMBD];
    if (threadIdx.x < EMBD) { ssum[threadIdx.x] = 0.0f; ssq[threadIdx.x] = 0.0f; }
    __syncthreads();

    const int dim    = threadIdx.x & 15;   // stride multiple of 16 -> dim fixed
    const int total  = N_NODES * EMBD;
    const int stride = gridDim.x * blockDim.x;
    float s = 0.0f, s2 = 0.0f;
    for (int i = blockIdx.x * blockDim.x + threadIdx.x; i < total; i += stride) {
        const float v = h[i];
        s  += v;
        s2 += v * v;
    }
    atomicAdd(&ssum[dim], s);              // ds_add_f32
    atomicAdd(&ssq[dim],  s2);
    __syncthreads();

    if (threadIdx.x < EMBD) {
        unsafeAtomicAdd(&stats[threadIdx.x],        ssum[threadIdx.x]);
        unsafeAtomicAdd(&stats[EMBD + threadIdx.x], ssq[threadIdx.x]);
    }
}

__global__ void zero_stats_kernel(float* __restrict__ stats) {
    if (threadIdx.x < 2 * EMBD) stats[threadIdx.x] = 0.0f;
}

// ---------------------------------------------------------------------------
// Phase 4: normalize + affine + ReLU + residual, in place on d_out.
// ---------------------------------------------------------------------------
__global__ void finalize_kernel(const float* __restrict__ x,
                                const float* __restrict__ stats,
                                const float* __restrict__ gamma,
                                const float* __restrict__ beta,
                                float* __restrict__ out) {
    const int idx = blockIdx.x * blockDim.x + threadIdx.x;
    if (idx >= N_NODES * EMBD) return;
    const int dim = idx & 15;
    const float invN = 1.0f / (float)N_NODES;
    const float mean = stats[dim] * invN;
    const float var  = stats[EMBD + dim] * invN - mean * mean;  // biased variance
    const float hn   = (out[idx] - mean) * rsqrtf(var + BN_EPS) * gamma[dim] + beta[dim];
    out[idx] = x[idx] + fmaxf(hn, 0.0f);
}

// ---------------------------------------------------------------------------
extern "C" void kernel_launch(void* const* d_in, const int* in_sizes, int n_in,
                              void* d_out, int out_size, void* d_ws, size_t ws_size,
                              hipStream_t stream) {
    const float* x         = (const float*)d_in[0];
    const int*   ei        = (const int*)  d_in[1];   // [2, N_EDGES]: src row then tgt row
    const float* edge_attr = (const float*)d_in[2];
    const float* Wk   = (const float*)d_in[3];
    const float* bk   = (const float*)d_in[4];
    const float* Wq   = (const float*)d_in[5];
    const float* bq   = (const float*)d_in[6];
    const float* Wv   = (const float*)d_in[7];
    const float* bv   = (const float*)d_in[8];
    const float* We   = (const float*)d_in[9];
    const float* Ws   = (const float*)d_in[10];
    const float* bs   = (const float*)d_in[11];
    const float* bias = (const float*)d_in[12];
    const float* gamma = (const float*)d_in[13];
    const float* beta  = (const float*)d_in[14];

    float* out = (float*)d_out;             // doubles as the h accumulation buffer

    // Workspace: k, q, v (32 MB each, L2-resident) + 32 floats of BN stats.
    float* kArr  = (float*)d_ws;
    float* qArr  = kArr + (size_t)N_NODES * EMBD;
    float* vArr  = qArr + (size_t)N_NODES * EMBD;
    float* stats = vArr + (size_t)N_NODES * EMBD;

    zero_stats_kernel<<<1, 64, 0, stream>>>(stats);

    const int nodeBlocks = (NODE_TILES + WAVES_PER_BLOCK - 1) / WAVES_PER_BLOCK;
    node_linear_kernel<<<nodeBlocks, 32 * WAVES_PER_BLOCK, 0, stream>>>(
        x, Wk, bk, Wq, bq, Wv, bv, Ws, bs, bias, kArr, qArr, vArr, out);

    const int edgeBlocks = (EDGE_TILES + WAVES_PER_BLOCK - 1) / WAVES_PER_BLOCK;
    edge_message_kernel<<<edgeBlocks, 32 * WAVES_PER_BLOCK, 0, stream>>>(
        edge_attr, ei, ei + N_EDGES, We, kArr, qArr, vArr, out);

    bn_stats_kernel<<<1024, 256, 0, stream>>>(out, stats);

    const int total = N_NODES * EMBD;
    finalize_kernel<<<(total + 255) / 256, 256, 0, stream>>>(x, stats, gamma, beta, out);
}